// MultiGraphConvolution_Layer_28656021799308
// MI455X (gfx1250) — compile-verified
//
#include <hip/hip_runtime.h>
#include <hip/hip_bf16.h>
#include <stdint.h>

#define N_NODES 50000
#define IN_F 128
#define OUT_F 64
#define NEG_SLOPE 0.2f

typedef __attribute__((ext_vector_type(2))) float v2f;
typedef __attribute__((ext_vector_type(8))) float v8f;

// ---------------------------------------------------------------------------
// Kernel 0: init out = 0, amax_key = 0 (== -inf under monotone key), denom = 0
// ---------------------------------------------------------------------------
__global__ __launch_bounds__(256) void gat_init(float* __restrict__ out,
                                                unsigned* __restrict__ amax_key,
                                                float* __restrict__ denom,
                                                int n_out, int n_nodes) {
    int i = blockIdx.x * blockDim.x + threadIdx.x;
    if (i < n_out) out[i] = 0.0f;
    if (i < n_nodes) { amax_key[i] = 0u; denom[i] = 0.0f; }
}

// ---------------------------------------------------------------------------
// Kernel 1: h = x @ W via V_WMMA_F32_16X16X4_F32, fused a_src/a_dst epilogue.
// 8 waves / block, each wave owns one 16-row tile, full 64 output columns.
// W is staged in LDS PRE-SWIZZLED into WMMA B-fragment order so each fragment
// is one aligned ds_load_b64 (conflict-free: lane*8B covers all 64 banks).
//
// Swizzled layout: Wl[ ((step*4 + nt)*32 + lane)*2 + j ]
//   = W[ (step*4 + khalf + j)*OUT_F + nt*16 + l15 ]
// where step = kk/4, khalf = (lane>>4)*2, l15 = lane&15, j in {0,1}.
// ---------------------------------------------------------------------------
__global__ __launch_bounds__(256) void gat_gemm(const float* __restrict__ x,
                                                const float* __restrict__ W,
                                                const float* __restrict__ att_src,
                                                const float* __restrict__ att_dst,
                                                float* __restrict__ h,
                                                float* __restrict__ a_src,
                                                float* __restrict__ a_dst) {
    __shared__ float Wl[IN_F * OUT_F];   // 32 KB, fragment-order swizzled
    __shared__ float att_s[OUT_F];
    __shared__ float att_d[OUT_F];

    const int tid = threadIdx.x;
    // Stage W swizzled: i = ((step*4 + nt)*32 + ln)*2 + j
    for (int i = tid; i < IN_F * OUT_F; i += 256) {
        const int j    = i & 1;
        const int ln   = (i >> 1) & 31;
        const int nt   = (i >> 6) & 3;
        const int step = i >> 8;
        const int khalf = (ln >> 4) * 2;
        const int l15   = ln & 15;
        Wl[i] = W[(step * 4 + khalf + j) * OUT_F + nt * 16 + l15];
    }
    if (tid < OUT_F) { att_s[tid] = att_src[tid]; att_d[tid] = att_dst[tid]; }
    __syncthreads();

    const int wave = tid >> 5;
    const int lane = tid & 31;
    const int row_tile = blockIdx.x * 8 + wave;          // 3125 tiles total
    if (row_tile >= N_NODES / 16) return;
    const int row_base = row_tile * 16;

    const int l15   = lane & 15;
    const int khalf = (lane >> 4) * 2;   // lanes 0-15: K 0,1 ; lanes 16-31: K 2,3
    const int rhalf = (lane >> 4) * 8;   // C/D: lanes 16-31 hold rows M=8..15

    v8f acc[4] = {v8f{}, v8f{}, v8f{}, v8f{}};

    const float* xrow = x + (size_t)(row_base + l15) * IN_F;
    const float* bbase = &Wl[lane * 2];  // per-lane fragment base

    for (int kk = 0; kk < IN_F; kk += 4) {
        v2f a;
        a.x = xrow[kk + khalf + 0];
        a.y = xrow[kk + khalf + 1];
        const int step = kk >> 2;
        #pragma unroll
        for (int nt = 0; nt < 4; ++nt) {
            // single aligned 8B LDS load per B fragment
            const v2f b = *(const v2f*)(bbase + (size_t)(step * 4 + nt) * 64);
            acc[nt] = __builtin_amdgcn_wmma_f32_16x16x4_f32(
                false, a, false, b, (short)0, acc[nt], false, false);
        }
    }

    // Epilogue: store h tile; fused attention dot products with shuffle reduce.
    #pragma unroll
    for (int r = 0; r < 8; ++r) {
        const int row = row_base + rhalf + r;
        float ps = 0.0f, pd = 0.0f;
        #pragma unroll
        for (int nt = 0; nt < 4; ++nt) {
            const float v  = acc[nt][r];
            const int col  = nt * 16 + l15;
            h[(size_t)row * OUT_F + col] = v;
            ps += v * att_s[col];
            pd += v * att_d[col];
        }
        // reduce over the 16 lanes of this half-wave
        #pragma unroll
        for (int off = 8; off > 0; off >>= 1) {
            ps += __shfl_xor(ps, off, 16);
            pd += __shfl_xor(pd, off, 16);
        }
        if (l15 == 0) { a_src[row] = ps; a_dst[row] = pd; }
    }
}

// ---------------------------------------------------------------------------
// Monotone float<->uint key so segment-max maps to native u32 atomic max.
// ---------------------------------------------------------------------------
__device__ __forceinline__ unsigned f2key(float f) {
    unsigned u = __float_as_uint(f);
    return (u & 0x80000000u) ? ~u : (u | 0x80000000u);
}
__device__ __forceinline__ float key2f(unsigned k) {
    unsigned u = (k & 0x80000000u) ? (k & 0x7FFFFFFFu) : ~k;
    return __uint_as_float(u);
}

__device__ __forceinline__ void edge_endpoints(const long long* __restrict__ ei,
                                               long long i, long long E,
                                               long long& s, long long& d) {
    if (i < E) { s = ei[i]; d = ei[E + i]; }
    else       { s = d = i - E; }          // implicit self-loops
}

// ---------------------------------------------------------------------------
// Kernel 2: segment max of leaky-relu'd attention logits
// ---------------------------------------------------------------------------
__global__ __launch_bounds__(256) void gat_edge_max(const long long* __restrict__ ei,
                                                    const float* __restrict__ a_src,
                                                    const float* __restrict__ a_dst,
                                                    unsigned* __restrict__ amax_key,
                                                    long long E, long long total) {
    long long i = (long long)blockIdx.x * blockDim.x + threadIdx.x;
    if (i >= total) return;
    long long s, d;
    edge_endpoints(ei, i, E, s, d);
    float e = a_src[s] + a_dst[d];
    e = (e > 0.0f) ? e : NEG_SLOPE * e;
    atomicMax(&amax_key[d], f2key(e));
}

// ---------------------------------------------------------------------------
// Kernel 3: p = exp(e - max[dst]); denom[dst] += p
// ---------------------------------------------------------------------------
__global__ __launch_bounds__(256) void gat_edge_exp(const long long* __restrict__ ei,
                                                    const float* __restrict__ a_src,
                                                    const float* __restrict__ a_dst,
                                                    const unsigned* __restrict__ amax_key,
                                                    float* __restrict__ p,
                                                    float* __restrict__ denom,
                                                    long long E, long long total) {
    long long i = (long long)blockIdx.x * blockDim.x + threadIdx.x;
    if (i >= total) return;
    long long s, d;
    edge_endpoints(ei, i, E, s, d);
    float e = a_src[s] + a_dst[d];
    e = (e > 0.0f) ? e : NEG_SLOPE * e;
    const float m  = key2f(amax_key[d]);
    const float pe = __expf(e - m);
    p[i] = pe;
    atomicAdd(&denom[d], pe);
}

// ---------------------------------------------------------------------------
// Kernel 4: out[dst] += (p/denom[dst]) * h[src]  — one wave per edge,
// 2 columns per lane: coalesced 256B gather + 64 non-returning f32 atomics.
// ---------------------------------------------------------------------------
__global__ __launch_bounds__(256) void gat_scatter(const long long* __restrict__ ei,
                                                   const float* __restrict__ p,
                                                   const float* __restrict__ denom,
                                                   const float* __restrict__ h,
                                                   float* __restrict__ out,
                                                   long long E, long long total) {
    const long long edge = (long long)blockIdx.x * 8 + (threadIdx.x >> 5);
    const int lane = threadIdx.x & 31;
    if (edge >= total) return;
    long long s, d;
    edge_endpoints(ei, edge, E, s, d);
    const float alpha = p[edge] / (denom[d] + 1e-16f);
    const float* hs = h + (size_t)s * OUT_F;
    float*       od = out + (size_t)d * OUT_F;
    atomicAdd(&od[lane],      alpha * hs[lane]);
    atomicAdd(&od[lane + 32], alpha * hs[lane + 32]);
}

// ---------------------------------------------------------------------------
// Kernel 5: out = relu(out + bias)
// ---------------------------------------------------------------------------
__global__ __launch_bounds__(256) void gat_finalize(float* __restrict__ out,
                                                    const float* __restrict__ bias,
                                                    int total) {
    int i = blockIdx.x * blockDim.x + threadIdx.x;
    if (i >= total) return;
    float v = out[i] + bias[i & (OUT_F - 1)];
    out[i] = v > 0.0f ? v : 0.0f;
}

// ---------------------------------------------------------------------------
extern "C" void kernel_launch(void* const* d_in, const int* in_sizes, int n_in,
                              void* d_out, int out_size, void* d_ws, size_t ws_size,
                              hipStream_t stream) {
    const float*     x        = (const float*)d_in[0];
    const long long* ei       = (const long long*)d_in[1];   // int64 per reference
    const float*     W        = (const float*)d_in[2];
    const float*     att_src  = (const float*)d_in[3];
    const float*     att_dst  = (const float*)d_in[4];
    const float*     bias     = (const float*)d_in[5];
    float*           out      = (float*)d_out;

    const long long E     = (long long)in_sizes[1] / 2;      // 800000
    const long long total = E + N_NODES;                     // +self-loops

    // workspace layout (all fits in ~17 MB)
    float*    h        = (float*)d_ws;                       // N*64
    float*    a_src    = h + (size_t)N_NODES * OUT_F;        // N
    float*    a_dst    = a_src + N_NODES;                    // N
    unsigned* amax_key = (unsigned*)(a_dst + N_NODES);       // N
    float*    denom    = (float*)(amax_key + N_NODES);       // N
    float*    p        = denom + N_NODES;                    // E+N

    const int n_out = N_NODES * OUT_F;

    // 0) zero accumulators
    gat_init<<<(n_out + 255) / 256, 256, 0, stream>>>(out, amax_key, denom, n_out, N_NODES);

    // 1) WMMA GEMM + attention coefficients
    const int n_tiles = N_NODES / 16;                        // 3125
    gat_gemm<<<(n_tiles + 7) / 8, 256, 0, stream>>>(x, W, att_src, att_dst, h, a_src, a_dst);

    // 2) segment max
    gat_edge_max<<<(int)((total + 255) / 256), 256, 0, stream>>>(ei, a_src, a_dst, amax_key, E, total);

    // 3) exp + denom
    gat_edge_exp<<<(int)((total + 255) / 256), 256, 0, stream>>>(ei, a_src, a_dst, amax_key, p, denom, E, total);

    // 4) weighted scatter-add, one wave per edge
    gat_scatter<<<(int)((total + 7) / 8), 256, 0, stream>>>(ei, p, denom, h, out, E, total);

    // 5) bias + relu
    gat_finalize<<<(n_out + 255) / 256, 256, 0, stream>>>(out, bias, n_out);
}